// BlockCompressLearnable_33389075759560
// MI455X (gfx1250) — compile-verified
//
#include <hip/hip_runtime.h>

// CDNA5 / gfx1250: wave32, WMMA 16x16x4 f32, async global->LDS loads.
typedef float v2f __attribute__((ext_vector_type(2)));
typedef float v8f __attribute__((ext_vector_type(8)));

#define DIM   8192
#define BS    64
#define NBLK  (DIM / BS)   // 128
#define LDT   68           // padded LDS row stride (floats); 68*4=272 bytes, 16B-aligned rows, bank-spread

__global__ __launch_bounds__(256)
void abw_block_tile_kernel(const float* __restrict__ W,
                           const float* __restrict__ Ab,   // [128][64][64]
                           const float* __restrict__ Bb,   // [128][64][64]
                           float* __restrict__ out)
{
    __shared__ float At[BS * LDT];
    __shared__ float Wt[BS * LDT];
    __shared__ float Bt[BS * LDT];
    __shared__ float Tt[BS * LDT];

    const int m = blockIdx.x;   // B column-slab index
    const int n = blockIdx.y;   // A row-slab index
    const int t = threadIdx.x;  // 0..255 (8 wave32s)

    // ---------------- Stage A_n, W_tile, B_m into LDS (async, 16B per lane per op) ----------------
    {
        const int r  = t >> 2;          // row 0..63
        const int c0 = (t & 3) * 16;    // 16-float chunk start

        const float* gA = Ab + (size_t)n * (BS * BS) + (size_t)r * BS + c0;
        const float* gB = Bb + (size_t)m * (BS * BS) + (size_t)r * BS + c0;
        const float* gW = W  + (size_t)(n * BS + r) * DIM + (size_t)m * BS + c0;

        unsigned lA = (unsigned)(unsigned long long)&At[r * LDT + c0];
        unsigned lW = (unsigned)(unsigned long long)&Wt[r * LDT + c0];
        unsigned lB = (unsigned)(unsigned long long)&Bt[r * LDT + c0];

#pragma unroll
        for (int j = 0; j < 4; ++j) {
            asm volatile("global_load_async_to_lds_b128 %0, %1, off"
                         :: "v"(lA + 16u * j), "v"((unsigned long long)(gA + 4 * j)) : "memory");
            asm volatile("global_load_async_to_lds_b128 %0, %1, off"
                         :: "v"(lW + 16u * j), "v"((unsigned long long)(gW + 4 * j)) : "memory");
            asm volatile("global_load_async_to_lds_b128 %0, %1, off"
                         :: "v"(lB + 16u * j), "v"((unsigned long long)(gB + 4 * j)) : "memory");
        }
    }
    asm volatile("s_wait_asynccnt 0" ::: "memory");
    __syncthreads();

    // ---------------- Wave / lane decomposition ----------------
    const int wv    = t >> 5;        // wave 0..7
    const int lane  = t & 31;
    const int l16   = lane & 15;
    const int khalf = (lane < 16) ? 0 : 2;  // A frag: lanes 16-31 hold K=2,3
    const int rowD  = (lane < 16) ? 0 : 8;  // C/D frag: lanes 16-31 hold M=8..15

    const int mi  = wv & 3;          // M subtile 0..3
    const int nj0 = (wv >> 2) * 2;   // N subtiles {nj0, nj0+1}

    // ---------------- Matmul 1: T = A_n @ W_tile ----------------
    v8f acc0 = {};
    v8f acc1 = {};
#pragma unroll
    for (int k = 0; k < BS; k += 4) {
        v2f a = *(const v2f*)&At[(mi * 16 + l16) * LDT + k + khalf];
        v2f b0, b1;
        b0.x = Wt[(k + khalf + 0) * LDT + (nj0 + 0) * 16 + l16];
        b0.y = Wt[(k + khalf + 1) * LDT + (nj0 + 0) * 16 + l16];
        b1.x = Wt[(k + khalf + 0) * LDT + (nj0 + 1) * 16 + l16];
        b1.y = Wt[(k + khalf + 1) * LDT + (nj0 + 1) * 16 + l16];
        acc0 = __builtin_amdgcn_wmma_f32_16x16x4_f32(false, a, false, b0, (short)0, acc0, false, false);
        acc1 = __builtin_amdgcn_wmma_f32_16x16x4_f32(false, a, false, b1, (short)0, acc1, false, false);
    }
#pragma unroll
    for (int v = 0; v < 8; ++v) {
        Tt[(mi * 16 + rowD + v) * LDT + (nj0 + 0) * 16 + l16] = acc0[v];
        Tt[(mi * 16 + rowD + v) * LDT + (nj0 + 1) * 16 + l16] = acc1[v];
    }
    __syncthreads();

    // ---------------- Matmul 2: out_tile = T @ B_m ----------------
    acc0 = (v8f){};
    acc1 = (v8f){};
#pragma unroll
    for (int k = 0; k < BS; k += 4) {
        v2f a = *(const v2f*)&Tt[(mi * 16 + l16) * LDT + k + khalf];
        v2f b0, b1;
        b0.x = Bt[(k + khalf + 0) * LDT + (nj0 + 0) * 16 + l16];
        b0.y = Bt[(k + khalf + 1) * LDT + (nj0 + 0) * 16 + l16];
        b1.x = Bt[(k + khalf + 0) * LDT + (nj0 + 1) * 16 + l16];
        b1.y = Bt[(k + khalf + 1) * LDT + (nj0 + 1) * 16 + l16];
        acc0 = __builtin_amdgcn_wmma_f32_16x16x4_f32(false, a, false, b0, (short)0, acc0, false, false);
        acc1 = __builtin_amdgcn_wmma_f32_16x16x4_f32(false, a, false, b1, (short)0, acc1, false, false);
    }

    // ---------------- Store output tile (coalesced: consecutive lanes -> consecutive columns) ----------------
    float* o = out + (size_t)(n * BS + mi * 16 + rowD) * DIM + (size_t)m * BS + l16;
#pragma unroll
    for (int v = 0; v < 8; ++v) {
        o[(size_t)v * DIM + (nj0 + 0) * 16] = acc0[v];
        o[(size_t)v * DIM + (nj0 + 1) * 16] = acc1[v];
    }
}

extern "C" void kernel_launch(void* const* d_in, const int* in_sizes, int n_in,
                              void* d_out, int out_size, void* d_ws, size_t ws_size,
                              hipStream_t stream) {
    (void)in_sizes; (void)n_in; (void)out_size; (void)d_ws; (void)ws_size;
    const float* W  = (const float*)d_in[0];
    const float* Ab = (const float*)d_in[1];
    const float* Bb = (const float*)d_in[2];
    float* out = (float*)d_out;

    dim3 grid(NBLK, NBLK);   // (m, n) = 128 x 128 output tiles of 64x64
    dim3 block(256);         // 8 wave32s
    abw_block_tile_kernel<<<grid, block, 0, stream>>>(W, Ab, Bb, out);
}